// AtomToTokenEncoder_35983236006598
// MI455X (gfx1250) — compile-verified
//
#include <hip/hip_runtime.h>
#include <hip/hip_bf16.h>

// ---------------------------------------------------------------------------
// AtomToTokenEncoder for MI455X (gfx1250, wave32).
// fp32 end-to-end using V_WMMA_F32_16X16X4_F32 for every dense GEMM.
// Attention exploits the sorted token_idx => block-diagonal mask.
// GEMM: 4 M-subtiles per wave (4 independent WMMA chains, 4x B reuse),
// compile-time N/K => immediate-offset addressing, float4 A staging.
// ---------------------------------------------------------------------------

typedef float v2f __attribute__((ext_vector_type(2)));
typedef float v8f __attribute__((ext_vector_type(8)));

static constexpr int N_ATOM   = 6144;
static constexpr int D_ATOM   = 128;
static constexpr int H        = 4;
static constexpr int DH       = 32;
static constexpr int D_FF     = 512;
static constexpr int D_MODEL  = 512;
static constexpr int N_PAIRS  = 49152;
static constexpr int BMAX     = 64;     // max atoms/token handled for pair bias
                                        // (expected max ~22 for Poisson(8) over 768 tokens)

// ------------------------------ utility kernels ----------------------------

__global__ void k_zero(float* __restrict__ p, int n) {
  int i = blockIdx.x * blockDim.x + threadIdx.x;
  if (i < n) p[i] = 0.0f;
}

// LayerNorm over last dim (d = 128), one row per 128-thread block.
__global__ __launch_bounds__(128) void k_layernorm(
    const float* __restrict__ x, const float* __restrict__ g,
    const float* __restrict__ b, float* __restrict__ y) {
  __shared__ float s1[128];
  __shared__ float s2[128];
  const int row = blockIdx.x;
  const int tid = threadIdx.x;
  float v = x[row * D_ATOM + tid];
  s1[tid] = v;
  s2[tid] = v * v;
  __syncthreads();
  for (int off = 64; off > 0; off >>= 1) {
    if (tid < off) { s1[tid] += s1[tid + off]; s2[tid] += s2[tid + off]; }
    __syncthreads();
  }
  const float mu  = s1[0] * (1.0f / D_ATOM);
  const float var = s2[0] * (1.0f / D_ATOM) - mu * mu;
  const float rs  = rsqrtf(var + 1e-5f);
  y[row * D_ATOM + tid] = (v - mu) * rs * g[tid] + b[tid];
}

// ------------------------------ WMMA f32 GEMM ------------------------------
// C[M,N] = A[M,K] * B[K,N]  (*sigmoid(gate)) (+bias[N]) (+res), all row-major.
// N, K compile-time => B/A offsets become instruction immediates.
// Block: 256 threads = 8 waves. Each wave owns a 64x16 C macro-tile
// (TM=4 independent 16x16 WMMA accumulator chains); block covers 64x128 of C.
// A chunk (64 x 64) staged in LDS via float4; stride 68 floats keeps
// ds_store_b128 16B-aligned and the v2f fragment reads bank-conflict-free.

static constexpr int LDS_K      = 64;
static constexpr int LDS_STRIDE = 68;
static constexpr int TM         = 4;        // 16-row subtiles per wave
static constexpr int BLK_M      = 16 * TM;  // 64 rows per block

template <int N, int K>
__global__ __launch_bounds__(256) void k_gemm_f32(
    const float* __restrict__ A, const float* __restrict__ B, float* __restrict__ C,
    const float* __restrict__ bias, const float* __restrict__ gate,
    const float* __restrict__ res) {
  __shared__ float As[BLK_M * LDS_STRIDE];
  const int lane = threadIdx.x & 31;
  const int wave = threadIdx.x >> 5;
  const int m0   = blockIdx.x * BLK_M;
  const int n0   = blockIdx.y * 128 + wave * 16;
  const int arow = lane & 15;   // A row within subtile / C column-lane
  const int asel = lane >> 4;   // K-pair select (ISA A 16x4 layout)

  v8f acc[TM];
#pragma unroll
  for (int mt = 0; mt < TM; ++mt) acc[mt] = (v8f){0.f,0.f,0.f,0.f,0.f,0.f,0.f,0.f};

  for (int kc = 0; kc < K; kc += LDS_K) {
    __syncthreads();
    // Cooperative float4 load of A[m0..m0+63][kc..kc+63] into LDS.
#pragma unroll
    for (int t = 0; t < (BLK_M * LDS_K) / (256 * 4); ++t) {
      const int idx = threadIdx.x + t * 256;      // float4 index
      const int r  = idx >> 4;                    // 16 float4 per row
      const int c4 = (idx & 15) << 2;
      const float4 v = *reinterpret_cast<const float4*>(&A[(m0 + r) * K + kc + c4]);
      *reinterpret_cast<float4*>(&As[r * LDS_STRIDE + c4]) = v;
    }
    if (kc + LDS_K < K)  // hint next A chunk into cache (global_prefetch_b8)
      __builtin_prefetch(&A[m0 * K + kc + LDS_K], 0, 0);
    __syncthreads();

    const float* Bp = B + kc * N + n0 + arow;     // per-chunk base; rest immediates
#pragma unroll
    for (int kk = 0; kk < LDS_K; kk += 4) {
      // B frag loaded once, reused by all TM subtiles.
      v2f b;
      b.x = Bp[(kk + asel * 2) * N];
      b.y = Bp[(kk + asel * 2 + 1) * N];
#pragma unroll
      for (int mt = 0; mt < TM; ++mt) {
        // A frag: lanes 0-15 hold K=k..k+1, lanes 16-31 hold K=k+2..k+3
        const v2f a = *(const v2f*)&As[(mt * 16 + arow) * LDS_STRIDE + kk + asel * 2];
        acc[mt] = __builtin_amdgcn_wmma_f32_16x16x4_f32(
            /*neg_a=*/false, a, /*neg_b=*/false, b,
            /*c_mod=*/(short)0, acc[mt], /*reuse_a=*/false, /*reuse_b=*/false);
      }
    }
  }

#pragma unroll
  for (int mt = 0; mt < TM; ++mt) {
#pragma unroll
    for (int r = 0; r < 8; ++r) {
      const int mr = m0 + mt * 16 + r + asel * 8;  // C layout: VGPR r -> rows r, r+8
      const int nc = n0 + arow;
      const int off = mr * N + nc;
      float v = acc[mt][r];
      if (gate) v *= 1.0f / (1.0f + __expf(-gate[off]));
      if (bias) v += bias[nc];
      if (res)  v += res[off];
      C[off] = v;
    }
  }
}

// --------------------------- token ranges (sorted) -------------------------

__global__ void k_token_ranges(const int* __restrict__ tok, int n_atom, int n_tokens,
                               int* __restrict__ start, int* __restrict__ cnt) {
  const int t = blockIdx.x * blockDim.x + threadIdx.x;
  if (t >= n_tokens) return;
  int lo = 0, hi = n_atom;
  while (lo < hi) { const int mid = (lo + hi) >> 1; if (tok[mid] < t) lo = mid + 1; else hi = mid; }
  const int s = lo;
  hi = n_atom;
  while (lo < hi) { const int mid = (lo + hi) >> 1; if (tok[mid] < t + 1) lo = mid + 1; else hi = mid; }
  start[t] = s;
  cnt[t]   = lo - s;
}

// --------------------------- pair bias + scatter ---------------------------

__global__ void k_pair_bias(const float* __restrict__ p_lm, const float* __restrict__ w_pb,
                            const float* __restrict__ b_pb, float* __restrict__ bh) {
  const int p = blockIdx.x * blockDim.x + threadIdx.x;
  if (p >= N_PAIRS) return;
  float f[16];
#pragma unroll
  for (int k = 0; k < 16; ++k) f[k] = p_lm[p * 16 + k];
#pragma unroll
  for (int h = 0; h < H; ++h) {
    float s = b_pb[h];
#pragma unroll
    for (int k = 0; k < 16; ++k) s += f[k] * w_pb[k * H + h];
    bh[p * H + h] = s;
  }
}

// dense[h][i][lj] = bias (only pairs inside a token block matter; others masked)
__global__ void k_scatter_bias(const int* __restrict__ idx, const int* __restrict__ tok,
                               const int* __restrict__ start, const float* __restrict__ bh,
                               float* __restrict__ dense) {
  const int p = blockIdx.x * blockDim.x + threadIdx.x;
  if (p >= N_PAIRS) return;
  const int i = idx[2 * p], j = idx[2 * p + 1];
  const int ti = tok[i];
  if (tok[j] != ti) return;
  const int lj = j - start[ti];
  if ((unsigned)lj >= (unsigned)BMAX) return;
#pragma unroll
  for (int h = 0; h < H; ++h)
    dense[(h * N_ATOM + i) * BMAX + lj] = bh[p * H + h];
}

// --------------------------- block-diagonal attention ----------------------
// One block per token; thread = (head, query). Blocks average ~8 atoms.

__global__ __launch_bounds__(256) void k_attention(
    const float* __restrict__ Q, const float* __restrict__ Km,
    const float* __restrict__ V, const float* __restrict__ dense,
    const int* __restrict__ start, const int* __restrict__ cnt,
    float* __restrict__ out) {
  const int t = blockIdx.x;
  const int s = start[t];
  const int n = cnt[t];
  if (n <= 0) return;
  const int h = threadIdx.x >> 6;  // 0..3
  const float scale = 0.17677669529663687f;  // 1/sqrt(32)

  for (int il = threadIdx.x & 63; il < n; il += 64) {
    const int i = s + il;
    float qv[DH];
    const float* qp = Q + i * D_ATOM + h * DH;
#pragma unroll
    for (int d = 0; d < DH; ++d) qv[d] = qp[d];
    const float* dp = dense + (h * N_ATOM + i) * BMAX;

    // pass 1: row max
    float m = -3.0e38f;
    for (int j = 0; j < n; ++j) {
      const float* kp = Km + (s + j) * D_ATOM + h * DH;
      float sc = 0.f;
#pragma unroll
      for (int d = 0; d < DH; ++d) sc += qv[d] * kp[d];
      sc = sc * scale + (j < BMAX ? dp[j] : 0.f);
      m = fmaxf(m, sc);
    }
    // pass 2: exp-sum + weighted V
    float denom = 0.f;
    float accv[DH];
#pragma unroll
    for (int d = 0; d < DH; ++d) accv[d] = 0.f;
    for (int j = 0; j < n; ++j) {
      const float* kp = Km + (s + j) * D_ATOM + h * DH;
      float sc = 0.f;
#pragma unroll
      for (int d = 0; d < DH; ++d) sc += qv[d] * kp[d];
      sc = sc * scale + (j < BMAX ? dp[j] : 0.f);
      const float p = __expf(sc - m);
      denom += p;
      const float* vp = V + (s + j) * D_ATOM + h * DH;
#pragma unroll
      for (int d = 0; d < DH; ++d) accv[d] += p * vp[d];
    }
    const float inv = 1.0f / denom;
    float* op = out + i * D_ATOM + h * DH;
#pragma unroll
    for (int d = 0; d < DH; ++d) op[d] = accv[d] * inv;
  }
}

// ------------------------------ SwiGLU combine -----------------------------

__global__ void k_swiglu(float* __restrict__ u1, const float* __restrict__ u2, int n) {
  const int i = blockIdx.x * blockDim.x + threadIdx.x;
  if (i >= n) return;
  const float a = u1[i];
  u1[i] = (a / (1.0f + __expf(-a))) * u2[i];
}

// ------------------------------ segment mean -------------------------------

__global__ __launch_bounds__(256) void k_segmean(
    const float* __restrict__ af, const int* __restrict__ start,
    const int* __restrict__ cnt, float* __restrict__ out) {
  const int t = blockIdx.x;
  const int s = start[t];
  const int n = cnt[t];
  const float inv = 1.0f / fmaxf((float)n, 1.0f);
  for (int c = threadIdx.x; c < D_MODEL; c += blockDim.x) {
    float sum = 0.f;
    for (int k = 0; k < n; ++k) sum += af[(s + k) * D_MODEL + c];
    out[t * D_MODEL + c] = sum * inv;
  }
}

// ------------------------------- launch glue -------------------------------

extern "C" void kernel_launch(void* const* d_in, const int* in_sizes, int n_in,
                              void* d_out, int out_size, void* d_ws, size_t ws_size,
                              hipStream_t stream) {
  const float* c_atom    = (const float*)d_in[0];
  const float* p_lm      = (const float*)d_in[1];
  const int*   p_lm_idx  = (const int*)d_in[2];
  const int*   token_idx = (const int*)d_in[3];
  // d_in[4] = n_tokens scalar (device); derive on host from out_size instead
  const float* ln_attn_g = (const float*)d_in[5];
  const float* ln_attn_b = (const float*)d_in[6];
  const float* w_q       = (const float*)d_in[7];
  const float* w_k       = (const float*)d_in[8];
  const float* w_v       = (const float*)d_in[9];
  const float* w_g       = (const float*)d_in[10];
  const float* w_o       = (const float*)d_in[11];
  const float* w_pb      = (const float*)d_in[12];
  const float* b_pb      = (const float*)d_in[13];
  const float* ln_ff_g   = (const float*)d_in[14];
  const float* ln_ff_b   = (const float*)d_in[15];
  const float* w1        = (const float*)d_in[16];
  const float* w2        = (const float*)d_in[17];
  const float* w3        = (const float*)d_in[18];
  const float* w_tok     = (const float*)d_in[19];
  const float* b_tok     = (const float*)d_in[20];
  float* out = (float*)d_out;

  const int n_tokens = out_size / D_MODEL;  // 768

  // ---- workspace layout (floats), aliased across pipeline phases ----
  float* W = (float*)d_ws;
  const size_t SZ_NM  = (size_t)N_ATOM * D_ATOM;   // 786432
  const size_t SZ_NF  = (size_t)N_ATOM * D_FF;     // 3145728
  float* qn    = W;                      // LN1 out; later LN2 out (hn)
  float* Qb    = W + SZ_NM;              // Q,K,V,G contiguous; later u1; later atom_feat
  float* Kb    = Qb + SZ_NM;
  float* Vb    = Kb + SZ_NM;
  float* Gb    = Vb + SZ_NM;
  float* attb  = Gb + SZ_NM;             // att_out; later q2
  float* q1    = attb + SZ_NM;           // post-attention residual state
  float* bh    = q1 + SZ_NM;             // pair bias (P x H)
  int*   rng   = (int*)(bh + (size_t)N_PAIRS * H);
  int*   startp = rng;                   // 768 ints
  int*   cntp   = rng + 1024;            // 768 ints
  float* dense = (float*)(rng + 2048);   // H x N_ATOM x BMAX; later overlapped by u2
  float* u1    = Qb;                     // reuse QKVG region (dead after w_o gemm)
  float* u2    = dense;                  // reuse dense region (dead after attention)
  float* q2    = attb;                   // reuse att_out (dead after w_o gemm)
  float* af    = Qb;                     // atom_feat reuses u1 (dead after w3 gemm)

  const int denseN = H * N_ATOM * BMAX;  // 1,572,864 floats

  // 1) zero dense pair-bias buffer (assignment semantics: unset pairs = 0)
  k_zero<<<(denseN + 255) / 256, 256, 0, stream>>>(dense, denseN);

  // 2) LN1
  k_layernorm<<<N_ATOM, 128, 0, stream>>>(c_atom, ln_attn_g, ln_attn_b, qn);

  // 3) Q/K/V/G projections (WMMA f32)
  {
    dim3 grid(N_ATOM / BLK_M, 1);
    k_gemm_f32<128, 128><<<grid, 256, 0, stream>>>(qn, w_q, Qb, nullptr, nullptr, nullptr);
    k_gemm_f32<128, 128><<<grid, 256, 0, stream>>>(qn, w_k, Kb, nullptr, nullptr, nullptr);
    k_gemm_f32<128, 128><<<grid, 256, 0, stream>>>(qn, w_v, Vb, nullptr, nullptr, nullptr);
    k_gemm_f32<128, 128><<<grid, 256, 0, stream>>>(qn, w_g, Gb, nullptr, nullptr, nullptr);
  }

  // 4) token ranges (sorted token_idx -> binary search, no atomics)
  k_token_ranges<<<(n_tokens + 255) / 256, 256, 0, stream>>>(token_idx, N_ATOM, n_tokens, startp, cntp);

  // 5) pair bias projection + scatter into dense per-block buffer
  k_pair_bias<<<(N_PAIRS + 255) / 256, 256, 0, stream>>>(p_lm, w_pb, b_pb, bh);
  k_scatter_bias<<<(N_PAIRS + 255) / 256, 256, 0, stream>>>(p_lm_idx, token_idx, startp, bh, dense);

  // 6) block-diagonal attention (mask => only same-token pairs survive)
  k_attention<<<n_tokens, 256, 0, stream>>>(Qb, Kb, Vb, dense, startp, cntp, attb);

  // 7) gated output projection + residual: q1 = c_atom + sigmoid(G) * (att @ w_o)
  k_gemm_f32<128, 128><<<dim3(N_ATOM / BLK_M, 1), 256, 0, stream>>>(attb, w_o, q1,
                                                                    nullptr, Gb, c_atom);

  // 8) LN2 (reuses qn buffer)
  k_layernorm<<<N_ATOM, 128, 0, stream>>>(q1, ln_ff_g, ln_ff_b, qn);

  // 9) SwiGLU FFN
  {
    dim3 grid(N_ATOM / BLK_M, D_FF / 128);
    k_gemm_f32<512, 128><<<grid, 256, 0, stream>>>(qn, w1, u1, nullptr, nullptr, nullptr);
    k_gemm_f32<512, 128><<<grid, 256, 0, stream>>>(qn, w2, u2, nullptr, nullptr, nullptr);
  }
  k_swiglu<<<(int)((SZ_NF + 255) / 256), 256, 0, stream>>>(u1, u2, (int)SZ_NF);
  // q2 = q1 + (silu(u1)*u2) @ w3
  k_gemm_f32<128, 512><<<dim3(N_ATOM / BLK_M, 1), 256, 0, stream>>>(u1, w3, q2,
                                                                    nullptr, nullptr, q1);

  // 10) token projection: af = q2 @ w_tok + b_tok
  k_gemm_f32<512, 128><<<dim3(N_ATOM / BLK_M, D_MODEL / 128), 256, 0, stream>>>(q2, w_tok, af,
                                                                                b_tok, nullptr, nullptr);

  // 11) segment mean over sorted token ranges
  k_segmean<<<n_tokens, 256, 0, stream>>>(af, startp, cntp, out);
}